// TreeCRF_44710609551704
// MI455X (gfx1250) — compile-verified
//
#include <hip/hip_runtime.h>
#include <hip/hip_bf16.h>
#include <math.h>

#define Bsz 32
#define Lsz 128
#define Dsz 400
#define Ksz 40
#define TLD 404   // padded LDS leading dim for the T tile
#define DCH 16    // d-rows per staged panel (4 WMMA k-steps)
#define NBLK 25   // 400 cols / 16
#define CPLD 20   // C panel leading dim (dwords): pair-aligned ds_load_b64

typedef __attribute__((ext_vector_type(2))) float v2f;
typedef __attribute__((ext_vector_type(8))) float v8f;

// ---------------------------------------------------------------------------
// Kernel 1: sd[b,k,h] = W_d[k]·H[b,h],  se[b,k,c] = W_e[k]·C[b,c]
// ---------------------------------------------------------------------------
__global__ __launch_bounds__(256) void sdse_kernel(
    const float* __restrict__ H, const float* __restrict__ C,
    const float* __restrict__ Wd, const float* __restrict__ We,
    float* __restrict__ sd, float* __restrict__ se) {
  int idx = blockIdx.x * blockDim.x + threadIdx.x;
  if (idx >= Bsz * Ksz * Lsz) return;
  int l = idx % Lsz;
  int k = (idx / Lsz) % Ksz;
  int b = idx / (Ksz * Lsz);
  const float4* hr = (const float4*)(H + ((size_t)b * Lsz + l) * Dsz);
  const float4* cr = (const float4*)(C + ((size_t)b * Lsz + l) * Dsz);
  const float4* wd = (const float4*)(Wd + (size_t)k * Dsz);
  const float4* we = (const float4*)(We + (size_t)k * Dsz);
  float s1 = 0.f, s2 = 0.f;
#pragma unroll 4
  for (int i = 0; i < Dsz / 4; ++i) {
    float4 hv = hr[i], wv = wd[i];
    s1 += hv.x * wv.x + hv.y * wv.y + hv.z * wv.z + hv.w * wv.w;
    float4 cv = cr[i], ev = we[i];
    s2 += cv.x * ev.x + cv.y * ev.y + cv.z * ev.z + cv.w * ev.w;
  }
  sd[idx] = s1;  // idx == (b*K + k)*L + l
  se[idx] = s2;
}

// ---------------------------------------------------------------------------
// Async global->LDS staging (CDNA5 ASYNCcnt path).
// ---------------------------------------------------------------------------
__device__ __forceinline__ void async_b128(unsigned lds_off, const float* gaddr) {
  asm volatile("global_load_async_to_lds_b128 %0, %1, off"
               :: "v"(lds_off), "v"(gaddr) : "memory");
}
__device__ __forceinline__ void wait_async0() {
  asm volatile("s_wait_asynccnt 0x0" ::: "memory");
}

// U panel, blocked layout [NBLK][DCH][16]: element (d, n) at
// (n/16)*256 + d*16 + n%16. Fragment pair (d, d+1) is 64B apart -> 2addr ds.
__device__ __forceinline__ void load_upanel(const float* __restrict__ Uk, int p,
                                            float* buf, int tid) {
  const float* src = Uk + (size_t)p * DCH * Dsz;
  for (int idx = tid; idx < DCH * (Dsz / 4); idx += 256) {
    const int r = idx / (Dsz / 4);        // d-row within panel
    const int j = idx - r * (Dsz / 4);    // float4 index along 400 cols
    const int nb = j >> 2;                // 16-col block
    const int nin = (j & 3) * 4;          // col within block
    async_b128((unsigned)(uintptr_t)(const void*)(buf + nb * (DCH * 16) + r * 16 + nin),
               src + r * Dsz + j * 4);
  }
}

// C panel: C_b[:, d0..d0+15] -> LDS [128][CPLD] (pairs contiguous for b64 reads)
__device__ __forceinline__ void load_cpanel(const float* __restrict__ Cb, int p,
                                            float* buf, int tid) {
  const int d0 = p * DCH;
  for (int idx = tid; idx < Lsz * 4; idx += 256) {
    const int c = idx >> 2;
    const int j = idx & 3;
    async_b128((unsigned)(uintptr_t)(const void*)(buf + c * CPLD + j * 4),
               Cb + (size_t)c * Dsz + d0 + j * 4);
  }
}

// ---------------------------------------------------------------------------
// Kernel 2: per (b,k) workgroup:  T = H_b @ U_k  (LDS),  E = T @ C_b^T,
// fused epilogue -> atomic accumulate exp-energies into A, gather tgt.
// ---------------------------------------------------------------------------
__global__ __launch_bounds__(256) void bilinear_kernel(
    const float* __restrict__ H, const float* __restrict__ C,
    const float* __restrict__ U, const float* __restrict__ bias,
    const float* __restrict__ mask, const int* __restrict__ heads,
    const int* __restrict__ tags, const float* __restrict__ sd,
    const float* __restrict__ se, float* __restrict__ Aout,
    float* __restrict__ tgt) {
  __shared__ float T[Lsz * TLD];               // 128 x 404 f32  (~202 KB)
  __shared__ float PanU[2][NBLK * DCH * 16];   // double-buffered U panel (50 KB)
  __shared__ float PanC[2][Lsz * CPLD];        // double-buffered C panel (20 KB)
  __shared__ float s_sd[Lsz], s_se[Lsz], s_mask[Lsz];
  __shared__ int s_heads[Lsz], s_tags[Lsz];

  const int bk = blockIdx.x;
  const int b = bk / Ksz;
  const int k = bk % Ksz;
  const int tid = threadIdx.x;
  const int wave = tid >> 5;
  const int lane = tid & 31;
  const int m0 = wave * 16;          // wave's output row block
  const int col = lane & 15;
  const int hi = lane >> 4;
  const int koff = hi * 2;           // A/B fragment K offset (16x4 f32 layout)
  const int rowoff = hi * 8;         // C/D fragment row offset
  const int row_a = m0 + col;

  if (tid < Lsz) {
    s_sd[tid]    = sd[(size_t)bk * Lsz + tid];
    s_se[tid]    = se[(size_t)bk * Lsz + tid];
    s_mask[tid]  = mask[(size_t)b * Lsz + tid];
    s_heads[tid] = heads[(size_t)b * Lsz + tid];
    s_tags[tid]  = tags[(size_t)b * Lsz + tid];
  }

  const float* Hb = H + (size_t)b * Lsz * Dsz;
  const float* Uk = U + (size_t)k * Dsz * Dsz;
  const float* Cb = C + (size_t)b * Lsz * Dsz;

  // warm caches for the H rows this workgroup streams (global_prefetch_b8)
  {
    const char* hb = (const char*)Hb;
    for (int i = tid; i < (Lsz * Dsz * 4) / 128; i += 256)
      __builtin_prefetch(hb + (size_t)i * 128, 0, 3);
  }

  const v8f vzero = {0.f, 0.f, 0.f, 0.f, 0.f, 0.f, 0.f, 0.f};

  // ======== Phase 1: T = Hb(128x400) @ Uk(400x400), U staged once via async ==
  v8f acc[NBLK];
#pragma unroll
  for (int t = 0; t < NBLK; ++t) acc[t] = vzero;

  auto compute_upanel = [&](const float* P, int p) {
#pragma unroll
    for (int ksl = 0; ksl < 4; ++ksl) {
      const int d0g = p * DCH + ksl * 4;
      const int dl = ksl * 4 + koff;
      v2f a = *(const v2f*)(Hb + (size_t)row_a * Dsz + d0g + koff);
#pragma unroll
      for (int t = 0; t < NBLK; ++t) {
        const float* Pt = P + t * (DCH * 16) + col;
        v2f bb;
        bb.x = Pt[dl * 16];        // (d, n)   -- base
        bb.y = Pt[dl * 16 + 16];   // (d+1, n) -- base + 64B (2addr-mergeable)
        acc[t] = __builtin_amdgcn_wmma_f32_16x16x4_f32(
            false, a, false, bb, (short)0, acc[t], false, false);
      }
    }
  };

  load_upanel(Uk, 0, PanU[0], tid);
  wait_async0();
  __syncthreads();

  for (int p = 0; p < NBLK; p += 2) {   // static double-buffer by parity
    if (p + 1 < NBLK) load_upanel(Uk, p + 1, PanU[1], tid);
    compute_upanel(PanU[0], p);
    wait_async0();
    __syncthreads();
    if (p + 1 < NBLK) {
      if (p + 2 < NBLK) load_upanel(Uk, p + 2, PanU[0], tid);
      compute_upanel(PanU[1], p + 1);
      wait_async0();
      __syncthreads();
    }
  }

#pragma unroll
  for (int t = 0; t < NBLK; ++t) {
    const int n0 = t * 16;
#pragma unroll
    for (int r = 0; r < 8; ++r)
      T[(m0 + r + rowoff) * TLD + n0 + col] = acc[t][r];
  }
  __syncthreads();

  // ======== Phase 2: E = T(128x400) @ Cb^T(400x128), C staged via async ======
  v8f acc2[8];
#pragma unroll
  for (int t = 0; t < 8; ++t) acc2[t] = vzero;

  auto compute_cpanel = [&](const float* P, int p) {
#pragma unroll
    for (int ksl = 0; ksl < 4; ++ksl) {
      const int d0g = p * DCH + ksl * 4;
      const int dl = ksl * 4 + koff;
      v2f a = *(const v2f*)(&T[row_a * TLD + d0g + koff]);       // ds_load_b64
#pragma unroll
      for (int t = 0; t < 8; ++t) {
        v2f bb = *(const v2f*)(&P[(t * 16 + col) * CPLD + dl]);  // ds_load_b64
        acc2[t] = __builtin_amdgcn_wmma_f32_16x16x4_f32(
            false, a, false, bb, (short)0, acc2[t], false, false);
      }
    }
  };

  load_cpanel(Cb, 0, PanC[0], tid);
  wait_async0();
  __syncthreads();

  for (int p = 0; p < NBLK; p += 2) {
    if (p + 1 < NBLK) load_cpanel(Cb, p + 1, PanC[1], tid);
    compute_cpanel(PanC[0], p);
    wait_async0();
    __syncthreads();
    if (p + 1 < NBLK) {
      if (p + 2 < NBLK) load_cpanel(Cb, p + 2, PanC[0], tid);
      compute_cpanel(PanC[1], p + 1);
      wait_async0();
      __syncthreads();
    }
  }

  // ======== Epilogue: mask/bias/diag/exp, accumulate A and tgt ==============
  const float bk_bias = bias[k];
#pragma unroll
  for (int t = 0; t < 8; ++t) {
    const int c = t * 16 + col;
#pragma unroll
    for (int r = 0; r < 8; ++r) {
      const int h = m0 + r + rowoff;
      const float m = s_mask[h] * s_mask[c];
      const float raw = acc2[t][r] + s_sd[h] + s_se[c] + bk_bias;
      const float em = raw * m;  // masked energy
      if (h != c && m != 0.0f)
        atomicAdd(&Aout[((size_t)b * Lsz + h) * Lsz + c], __expf(em));
      if (c >= 1 && s_heads[c] == h && s_tags[c] == k)
        atomicAdd(&tgt[b], em);
    }
  }
}

// ---------------------------------------------------------------------------
// Kernel 3: per batch: Laplacian submatrix, no-pivot LU in LDS (column
// diagonally dominant => stable, pivots > 0), out = logdet - tgt.
// ---------------------------------------------------------------------------
__global__ __launch_bounds__(128) void lu_kernel(
    const float* __restrict__ Aglob, const float* __restrict__ tgt,
    const int* __restrict__ lengths, float* __restrict__ out) {
  __shared__ float M[127 * 129];
  __shared__ float Dg[Lsz];
  __shared__ float red[Lsz];
  const int b = blockIdx.x;
  const int tid = threadIdx.x;
  const float* Ab = Aglob + (size_t)b * Lsz * Lsz;

  {  // Dg[c] = (sum_h A[h,c]) * (1+RTOL) + ATOL
    float s = 0.f;
    for (int h = 0; h < Lsz; ++h) s += Ab[h * Lsz + tid];
    Dg[tid] = s * (1.0f + 1.0e-4f) + 1.0e-6f;
  }
  __syncthreads();

  const int nv = lengths[b] - 1;
  if (tid < 127) {
    const int i = tid;
    const bool vi = i < nv;
    for (int j = 0; j < 127; ++j) {
      const bool v = vi && (j < nv);
      float val;
      if (v)
        val = ((i == j) ? Dg[j + 1] : 0.0f) - Ab[(i + 1) * Lsz + (j + 1)];
      else
        val = (i == j) ? 1.0f : 0.0f;
      M[i * 129 + j] = val;
    }
  }
  __syncthreads();

  for (int kk = 0; kk < 126; ++kk) {
    if (tid > kk && tid < 127) {
      const float f = M[tid * 129 + kk] / M[kk * 129 + kk];
      for (int j = kk + 1; j < 127; ++j)
        M[tid * 129 + j] -= f * M[kk * 129 + j];
    }
    __syncthreads();
  }

  red[tid] = (tid < 127) ? __logf(fabsf(M[tid * 129 + tid])) : 0.0f;
  __syncthreads();
  for (int s = 64; s > 0; s >>= 1) {
    if (tid < s) red[tid] += red[tid + s];
    __syncthreads();
  }
  if (tid == 0) out[b] = red[0] - tgt[b];
}

// ---------------------------------------------------------------------------
extern "C" void kernel_launch(void* const* d_in, const int* in_sizes, int n_in,
                              void* d_out, int out_size, void* d_ws, size_t ws_size,
                              hipStream_t stream) {
  const float* H       = (const float*)d_in[0];
  const float* C       = (const float*)d_in[1];
  const float* Wd      = (const float*)d_in[2];
  const float* We      = (const float*)d_in[3];
  const float* U       = (const float*)d_in[4];
  const float* bias    = (const float*)d_in[5];
  const float* mask    = (const float*)d_in[6];
  const int*   heads   = (const int*)d_in[7];
  const int*   tags    = (const int*)d_in[8];
  const int*   lengths = (const int*)d_in[9];
  float* out = (float*)d_out;

  float* sd  = (float*)d_ws;                          // B*K*L
  float* se  = sd + (size_t)Bsz * Ksz * Lsz;          // B*K*L
  float* Aws = se + (size_t)Bsz * Ksz * Lsz;          // B*L*L
  float* tgt = Aws + (size_t)Bsz * Lsz * Lsz;         // B

  hipMemsetAsync(Aws, 0, (size_t)Bsz * Lsz * Lsz * sizeof(float), stream);
  hipMemsetAsync(tgt, 0, (size_t)Bsz * sizeof(float), stream);

  sdse_kernel<<<(Bsz * Ksz * Lsz + 255) / 256, 256, 0, stream>>>(H, C, Wd, We, sd, se);
  bilinear_kernel<<<Bsz * Ksz, 256, 0, stream>>>(H, C, U, bias, mask, heads, tags,
                                                 sd, se, Aws, tgt);
  lu_kernel<<<Bsz, 128, 0, stream>>>(Aws, tgt, lengths, out);
}